// Gaussian_cross_attn_75342316306678
// MI455X (gfx1250) — compile-verified
//
#include <hip/hip_runtime.h>
#include <hip/hip_bf16.h>

typedef __attribute__((ext_vector_type(2))) float v2f;
typedef __attribute__((ext_vector_type(8))) float v8f;

// out[b,m,d] = sum_{k<5} w[b,m,k] * Q[b,k,d]
// w derived analytically from segment_len (Gaussian pdf, min-max normalized,
// masked for i >= n, special case n == 1 -> row 0 = ones).
//
// WMMA mapping (CDNA5, wave32): per 16(M) x 16(D) output tile,
//   A = 16x4 f32 slice of w (two k-slices: k=0..3 and k=4..7, k>4 zero)
//   B = 4x16 f32 slice of Q
//   two V_WMMA_F32_16X16X4_F32 accumulate the K=5 (padded to 8) reduction.
//
// Block = 128 threads = 4 waves. Each wave owns a 16-wide D chunk and loops
// over NT tiles (stride 64 columns), reusing the A (weight) fragments so the
// expf/normalize work is amortized NT x. All tiles are full (host guarantees
// divisibility), so the store path is guard-free straight-line code.

template <int NT>
__global__ __launch_bounds__(128) void gauss_xattn_wmma_kernel(
    const int*   __restrict__ seg_len,   // [B]
    const float* __restrict__ Q,         // [B,5,D]
    float*       __restrict__ out,       // [B,M,D]
    int M, int D)
{
  const int  lane = threadIdx.x & 31;
  const int  wave = threadIdx.x >> 5;
  const bool hi   = lane >= 16;
  const int  col  = lane & 15;

  const int b  = blockIdx.z;
  const int m0 = blockIdx.y << 4;
  const int d0 = blockIdx.x * (NT * 64) + (wave << 4);

  const int n    = seg_len[b];
  const int mrow = m0 + col;            // lanes 0-15 and 16-31 both cover M=0..15 of tile

  // ---- Gaussian weight row w[mrow][0..4], computed once per (b, m-tile) ----
  const float denom = (n > 1) ? (float)(n - 1) : 1.0f;
  const float mu    = (float)mrow * 4.0f / denom;

  float l[5];
#pragma unroll
  for (int k = 0; k < 5; ++k) {
    float z = ((float)k - mu) * (1.0f / 2.5f);
    l[k] = expf(-0.5f * z * z);
  }
  float lmin = l[0], lmax = l[0];
#pragma unroll
  for (int k = 1; k < 5; ++k) {
    lmin = fminf(lmin, l[k]);
    lmax = fmaxf(lmax, l[k]);
  }
  const float inv = 1.0f / (lmax - lmin);

  float w[5];
#pragma unroll
  for (int k = 0; k < 5; ++k) {
    float wk = (l[k] - lmin) * inv;
    wk = (mrow < n) ? wk : 0.0f;                       // zero invalid rows (kills NaN)
    wk = (n == 1) ? ((mrow == 0) ? 1.0f : 0.0f) : wk;  // n==1 special case
    w[k] = wk;
  }

  // ---- A fragments: 16x4 f32, VGPR0 = K(0|2), VGPR1 = K(1|3) over lane halves ----
  v2f a01, a45;
  a01.x = hi ? w[2] : w[0];
  a01.y = hi ? w[3] : w[1];
  a45.x = hi ? 0.0f : w[4];   // k=4 (k=6 slot is zero)
  a45.y = 0.0f;               // k=5 / k=7 are padding

  const size_t k0 = hi ? 2u : 0u;
  const float* Qb = Q + (size_t)b * 5u * (size_t)D + (size_t)(d0 + col);
  float*       ob = out + ((size_t)b * (size_t)M + (size_t)(m0 + (hi ? 8 : 0))) * (size_t)D
                        + (size_t)(d0 + col);

#pragma unroll
  for (int t = 0; t < NT; ++t) {
    // ---- B fragments: 4x16 f32, K striped across lane halves ----
    v2f b01, b45;
    b01.x = Qb[k0 * (size_t)D];
    b01.y = Qb[(k0 + 1u) * (size_t)D];
    float q4 = Qb[4u * (size_t)D];
    b45.x = hi ? 0.0f : q4;
    b45.y = 0.0f;

    // ---- D = A*B + C via two f32 WMMAs (EXEC all-1s throughout kernel) ----
    v8f c = {};
    c = __builtin_amdgcn_wmma_f32_16x16x4_f32(false, a01, false, b01,
                                              (short)0, c, false, false);
    c = __builtin_amdgcn_wmma_f32_16x16x4_f32(false, a45, false, b45,
                                              (short)0, c, false, false);

    // ---- Store 16x16 f32 C tile: VGPR r -> rows (r | r+8) over lane halves ----
#pragma unroll
    for (int r = 0; r < 8; ++r) {
      ob[(size_t)r * (size_t)D] = c[r];
    }

    Qb += 64;   // next 16-wide tile for this wave (4 waves cover 64 cols)
    ob += 64;
  }
}

extern "C" void kernel_launch(void* const* d_in, const int* in_sizes, int n_in,
                              void* d_out, int out_size, void* d_ws, size_t ws_size,
                              hipStream_t stream) {
  // Inputs (setup_inputs order): segment_len [B] int, max_chunk (scalar), Q_emb [B,5,D] f32
  const int*   seg = (const int*)d_in[0];
  const float* Q   = (const float*)d_in[2];
  float*       out = (float*)d_out;

  const int B = in_sizes[0];
  const int D = in_sizes[2] / (5 * B);
  const int M = out_size / (B * D);

  dim3 block(128, 1, 1);
  if ((D % 256) == 0 && (M % 16) == 0) {
    dim3 grid((unsigned)(D / 256), (unsigned)(M / 16), (unsigned)B);
    gauss_xattn_wmma_kernel<4><<<grid, block, 0, stream>>>(seg, Q, out, M, D);
  } else {
    dim3 grid((unsigned)(D / 64), (unsigned)(M / 16), (unsigned)B);
    gauss_xattn_wmma_kernel<1><<<grid, block, 0, stream>>>(seg, Q, out, M, D);
  }
}